// GRUVAE_566935683779
// MI455X (gfx1250) — compile-verified
//
#include <hip/hip_runtime.h>
#include <cstddef>

// ---------------- problem dims ----------------
#define INPUT_D   64
#define HIDDEN_D  512
#define LATENT_D  128
#define BATCH_D   64
#define T_STEPS   1024
#define RECON_N   (BATCH_D * T_STEPS * INPUT_D)   // 4194304

// ---------------- persistent-kernel config ----------------
#define NWG    16          // workgroups in the recurrent kernels (co-resident)
#define CHUNK  32          // hidden units owned per workgroup (512/16)
#define NLOC   (3*CHUNK)   // gate columns per workgroup = 96
#define NSTR   (NLOC*2)    // LDS row stride in floats for pair-interleaved weights
#define GSTR   33          // gate-staging LDS row stride (bank-conflict free)

typedef __attribute__((ext_vector_type(2))) float v2f;
typedef __attribute__((ext_vector_type(8))) float v8f;

// D(16x16,f32) = A(16x4,f32) x B(4x16,f32) + C  -- CDNA5 wave32 WMMA
__device__ __forceinline__ v8f wmma4(v2f a, v2f b, v8f c) {
  return __builtin_amdgcn_wmma_f32_16x16x4_f32(false, a, false, b, (short)0, c,
                                               false, false);
}

__device__ __forceinline__ float sigmoid_f(float x) {
  return 1.0f / (1.0f + expf(-x));
}

// h is stored pair-interleaved: h[k][b] lives at (k>>1)*128 + b*2 + (k&1)
__device__ __forceinline__ int hidx(int k, int b) {
  return ((k >> 1) * (2 * BATCH_D)) + b * 2 + (k & 1);
}

// Device-wide barrier: monotonic counter in ws, reset each launch by prep_kernel.
__device__ __forceinline__ void grid_barrier(unsigned int* cnt, unsigned int target) {
  __threadfence();
  __syncthreads();
  if (threadIdx.x == 0) {
    __hip_atomic_fetch_add(cnt, 1u, __ATOMIC_RELEASE, __HIP_MEMORY_SCOPE_AGENT);
    while (__hip_atomic_load(cnt, __ATOMIC_ACQUIRE, __HIP_MEMORY_SCOPE_AGENT) < target) {
      __builtin_amdgcn_s_sleep(2);
    }
  }
  __syncthreads();
  __threadfence();
}

// ---------------- K1: prep (zero h0, swizzle out_W, reset barrier counters) ----
__global__ __launch_bounds__(256) void prep_kernel(const float* __restrict__ outW,
                                                   float* __restrict__ hbuf0,
                                                   float* __restrict__ outWswz,
                                                   unsigned int* __restrict__ cnt) {
  int id = blockIdx.x * 256 + threadIdx.x;   // 0 .. 32767
  if (id < 2) cnt[id] = 0u;
  if (id < HIDDEN_D * BATCH_D) {
    hbuf0[id] = 0.0f;                        // encoder h0 = 0
    // outWswz[p][n][j] = out_W[n][2p+j]  (B-fragment friendly, pair-interleaved)
    int j = id & 1, n = (id >> 1) & 63, p = id >> 7;
    outWswz[id] = outW[n * HIDDEN_D + 2 * p + j];
  }
}

// ---------------- K2: encoder GRU scan (persistent, NWG workgroups) -----------
__global__ __launch_bounds__(256) void enc_kernel(
    const float* __restrict__ x, const float* __restrict__ Wih,
    const float* __restrict__ Whh, const float* __restrict__ bih,
    const float* __restrict__ bhh, float* __restrict__ hbuf0,
    float* __restrict__ hbuf1, unsigned int* __restrict__ cnt) {
  extern __shared__ float smem[];
  float* Whh_s  = smem;                                   // 256*NSTR = 49152
  float* Wih_s  = Whh_s + (HIDDEN_D / 2) * NSTR;          // 32*NSTR  = 6144
  float* g_r    = Wih_s + (INPUT_D / 2) * NSTR;           // 64*GSTR
  float* g_z    = g_r  + BATCH_D * GSTR;
  float* g_nh   = g_z  + BATCH_D * GSTR;
  float* g_nx   = g_nh + BATCH_D * GSTR;
  float* bsum_r = g_nx + BATCH_D * GSTR;                  // CHUNK each
  float* bsum_z = bsum_r + CHUNK;
  float* bx_n   = bsum_z + CHUNK;
  float* bh_n   = bx_n + CHUNK;

  const int tid = threadIdx.x;
  const int g = blockIdx.x;
  const int jbase = g * CHUNK;

  // ---- load this WG's weight slices into LDS (pair-interleaved) ----
  for (int idx = tid; idx < NLOC * HIDDEN_D; idx += 256) {
    int nl = idx / HIDDEN_D, k = idx % HIDDEN_D;
    int row = (nl / CHUNK) * HIDDEN_D + jbase + (nl % CHUNK);
    Whh_s[(k >> 1) * NSTR + nl * 2 + (k & 1)] = Whh[row * HIDDEN_D + k];
  }
  for (int idx = tid; idx < NLOC * INPUT_D; idx += 256) {
    int nl = idx / INPUT_D, k = idx % INPUT_D;
    int row = (nl / CHUNK) * HIDDEN_D + jbase + (nl % CHUNK);
    Wih_s[(k >> 1) * NSTR + nl * 2 + (k & 1)] = Wih[row * INPUT_D + k];
  }
  for (int jj = tid; jj < CHUNK; jj += 256) {
    int jg = jbase + jj;
    bsum_r[jj] = bih[jg] + bhh[jg];
    bsum_z[jj] = bih[HIDDEN_D + jg] + bhh[HIDDEN_D + jg];
    bx_n[jj]   = bih[2 * HIDDEN_D + jg];
    bh_n[jj]   = bhh[2 * HIDDEN_D + jg];
  }
  __syncthreads();

  const int wave = tid >> 5, lane = tid & 31;
  const int lo = lane & 15, hi = lane >> 4;
  const int mt = wave & 3, cg = wave >> 2;     // 4 M-tiles x 2 col-groups
  const int mrow = mt * 16 + lo;               // batch index for A fragments
  const int nr = (cg * 16 + lo) * 2;           // r-gate col (LDS float offset)
  const int nz = (32 + cg * 16 + lo) * 2;      // z-gate col
  const int nn = (64 + cg * 16 + lo) * 2;      // n-gate col

  for (int t = 0; t < T_STEPS; ++t) {
    const float* hc = (t & 1) ? hbuf1 : hbuf0;
    float* hn = (t & 1) ? hbuf0 : hbuf1;

    if (t + 1 < T_STEPS)
      __builtin_prefetch(x + ((size_t)mrow * T_STEPS + t + 1) * INPUT_D, 0, 0);

    v8f cr = {}, cz = {}, cnh = {}, cnx = {};
    // gates += h @ W_hh^T   (K = 512)
#pragma unroll 4
    for (int s = 0; s < HIDDEN_D / 4; ++s) {
      int p = 2 * s + hi;
      v2f a = *(const v2f*)(hc + (p * BATCH_D + mrow) * 2);
      const float* wrow = Whh_s + p * NSTR;
      v2f br = *(const v2f*)(wrow + nr);
      v2f bz = *(const v2f*)(wrow + nz);
      v2f bn = *(const v2f*)(wrow + nn);
      cr  = wmma4(a, br, cr);
      cz  = wmma4(a, bz, cz);
      cnh = wmma4(a, bn, cnh);
    }
    // gates += x_t @ W_ih^T (K = 64); n-gate x-part kept separate (cnx)
#pragma unroll
    for (int s = 0; s < INPUT_D / 4; ++s) {
      int p = 2 * s + hi;
      v2f a = *(const v2f*)(x + ((size_t)mrow * T_STEPS + t) * INPUT_D + p * 2);
      const float* wrow = Wih_s + p * NSTR;
      v2f br = *(const v2f*)(wrow + nr);
      v2f bz = *(const v2f*)(wrow + nz);
      v2f bn = *(const v2f*)(wrow + nn);
      cr  = wmma4(a, br, cr);
      cz  = wmma4(a, bz, cz);
      cnx = wmma4(a, bn, cnx);
    }
    // stage C fragments: (M = v + 8*hi within tile, N = lo)
    int cc = cg * 16 + lo;
#pragma unroll
    for (int v = 0; v < 8; ++v) {
      int b = mt * 16 + v + 8 * hi;
      g_r [b * GSTR + cc] = cr[v];
      g_z [b * GSTR + cc] = cz[v];
      g_nh[b * GSTR + cc] = cnh[v];
      g_nx[b * GSTR + cc] = cnx[v];
    }
    __syncthreads();
    // elementwise GRU update for this WG's 32 hidden units
#pragma unroll
    for (int u = 0; u < (BATCH_D * CHUNK) / 256; ++u) {
      int flat = u * 256 + tid;
      int jj = flat >> 6, b = flat & 63;
      int jg = jbase + jj;
      float r  = sigmoid_f(g_r[b * GSTR + jj] + bsum_r[jj]);
      float zz = sigmoid_f(g_z[b * GSTR + jj] + bsum_z[jj]);
      float n  = tanhf(g_nx[b * GSTR + jj] + bx_n[jj] +
                       r * (g_nh[b * GSTR + jj] + bh_n[jj]));
      float hold = hc[hidx(jg, b)];
      hn[hidx(jg, b)] = (1.0f - zz) * n + zz * hold;
    }
    grid_barrier(cnt, (unsigned int)(NWG * (t + 1)));
  }
}

// ---------------- K3a: mu / logvar heads + reparameterize ---------------------
__global__ __launch_bounds__(256) void latent_kernel(
    const float* __restrict__ eps, const float* __restrict__ muW,
    const float* __restrict__ muB, const float* __restrict__ lvW,
    const float* __restrict__ lvB, const float* __restrict__ hbuf0,
    float* __restrict__ zbuf, float* __restrict__ out) {
  int id = blockIdx.x * 256 + threadIdx.x;   // 0 .. 8191
  int b = id >> 7, l = id & 127;
  float am = 0.0f, av = 0.0f;
  for (int k = 0; k < HIDDEN_D; ++k) {
    float h = hbuf0[hidx(k, b)];
    am += h * muW[l * HIDDEN_D + k];
    av += h * lvW[l * HIDDEN_D + k];
  }
  float mu = am + muB[l];
  float lv = av + lvB[l];
  out[RECON_N + b * LATENT_D + l] = mu;
  out[RECON_N + BATCH_D * LATENT_D + b * LATENT_D + l] = lv;
  zbuf[b * LATENT_D + l] = mu + eps[b * LATENT_D + l] * expf(0.5f * lv);
}

// ---------------- K3b: h_dec = z @ l2h_W^T + b --------------------------------
__global__ __launch_bounds__(256) void l2h_kernel(const float* __restrict__ l2hW,
                                                  const float* __restrict__ l2hB,
                                                  const float* __restrict__ zbuf,
                                                  float* __restrict__ hbuf0) {
  int id = blockIdx.x * 256 + threadIdx.x;   // 0 .. 32767
  int b = id >> 9, j = id & 511;
  float a = 0.0f;
  for (int l = 0; l < LATENT_D; ++l)
    a += zbuf[b * LATENT_D + l] * l2hW[j * LATENT_D + l];
  hbuf0[hidx(j, b)] = a + l2hB[j];           // decoder initial h (buffer 0)
}

// ---------------- K4: decoder GRU scan with autoregressive feedback -----------
__global__ __launch_bounds__(256) void dec_kernel(
    const float* __restrict__ Wih, const float* __restrict__ Whh,
    const float* __restrict__ bih, const float* __restrict__ bhh,
    const float* __restrict__ outB, const float* __restrict__ outWswz,
    float* __restrict__ hbuf0, float* __restrict__ hbuf1,
    float* __restrict__ dout, unsigned int* __restrict__ cnt) {
  extern __shared__ float smem[];
  float* Whh_s  = smem;
  float* Wih_s  = Whh_s + (HIDDEN_D / 2) * NSTR;
  float* g_r    = Wih_s + (INPUT_D / 2) * NSTR;
  float* g_z    = g_r  + BATCH_D * GSTR;
  float* g_nh   = g_z  + BATCH_D * GSTR;
  float* g_nx   = g_nh + BATCH_D * GSTR;
  float* bsum_r = g_nx + BATCH_D * GSTR;
  float* bsum_z = bsum_r + CHUNK;
  float* bx_n   = bsum_z + CHUNK;
  float* bh_n   = bx_n + CHUNK;
  float* inpS   = bh_n + CHUNK;              // 32*64*2 = 4096 (pair-interleaved)
  float* ob_s   = inpS + (INPUT_D / 2) * BATCH_D * 2;  // 64

  const int tid = threadIdx.x;
  const int g = blockIdx.x;
  const int jbase = g * CHUNK;

  for (int idx = tid; idx < NLOC * HIDDEN_D; idx += 256) {
    int nl = idx / HIDDEN_D, k = idx % HIDDEN_D;
    int row = (nl / CHUNK) * HIDDEN_D + jbase + (nl % CHUNK);
    Whh_s[(k >> 1) * NSTR + nl * 2 + (k & 1)] = Whh[row * HIDDEN_D + k];
  }
  for (int idx = tid; idx < NLOC * INPUT_D; idx += 256) {
    int nl = idx / INPUT_D, k = idx % INPUT_D;
    int row = (nl / CHUNK) * HIDDEN_D + jbase + (nl % CHUNK);
    Wih_s[(k >> 1) * NSTR + nl * 2 + (k & 1)] = Wih[row * INPUT_D + k];
  }
  for (int jj = tid; jj < CHUNK; jj += 256) {
    int jg = jbase + jj;
    bsum_r[jj] = bih[jg] + bhh[jg];
    bsum_z[jj] = bih[HIDDEN_D + jg] + bhh[HIDDEN_D + jg];
    bx_n[jj]   = bih[2 * HIDDEN_D + jg];
    bh_n[jj]   = bhh[2 * HIDDEN_D + jg];
  }
  for (int i = tid; i < INPUT_D; i += 256) ob_s[i] = outB[i];
  for (int i = tid; i < INPUT_D * BATCH_D; i += 256) inpS[i] = 0.0f;  // x0 = 0
  __syncthreads();

  const int wave = tid >> 5, lane = tid & 31;
  const int lo = lane & 15, hi = lane >> 4;
  const int mt = wave & 3, cg = wave >> 2;
  const int mrow = mt * 16 + lo;
  const int nr = (cg * 16 + lo) * 2;
  const int nz = (32 + cg * 16 + lo) * 2;
  const int nn = (64 + cg * 16 + lo) * 2;
  const int mt2 = wave >> 1;                 // out-projection tiling
  const int nc = (wave & 1) * 32;

  for (int t = 0; t < T_STEPS; ++t) {
    const float* hc = (t & 1) ? hbuf1 : hbuf0;
    float* hn = (t & 1) ? hbuf0 : hbuf1;

    v8f cr = {}, cz = {}, cnh = {}, cnx = {};
#pragma unroll 4
    for (int s = 0; s < HIDDEN_D / 4; ++s) {
      int p = 2 * s + hi;
      v2f a = *(const v2f*)(hc + (p * BATCH_D + mrow) * 2);
      const float* wrow = Whh_s + p * NSTR;
      v2f br = *(const v2f*)(wrow + nr);
      v2f bz = *(const v2f*)(wrow + nz);
      v2f bn = *(const v2f*)(wrow + nn);
      cr  = wmma4(a, br, cr);
      cz  = wmma4(a, bz, cz);
      cnh = wmma4(a, bn, cnh);
    }
#pragma unroll
    for (int s = 0; s < INPUT_D / 4; ++s) {
      int p = 2 * s + hi;
      v2f a = *(const v2f*)(inpS + (p * BATCH_D + mrow) * 2);   // out_{t-1}
      const float* wrow = Wih_s + p * NSTR;
      v2f br = *(const v2f*)(wrow + nr);
      v2f bz = *(const v2f*)(wrow + nz);
      v2f bn = *(const v2f*)(wrow + nn);
      cr  = wmma4(a, br, cr);
      cz  = wmma4(a, bz, cz);
      cnx = wmma4(a, bn, cnx);
    }
    int cc = cg * 16 + lo;
#pragma unroll
    for (int v = 0; v < 8; ++v) {
      int b = mt * 16 + v + 8 * hi;
      g_r [b * GSTR + cc] = cr[v];
      g_z [b * GSTR + cc] = cz[v];
      g_nh[b * GSTR + cc] = cnh[v];
      g_nx[b * GSTR + cc] = cnx[v];
    }
    __syncthreads();
#pragma unroll
    for (int u = 0; u < (BATCH_D * CHUNK) / 256; ++u) {
      int flat = u * 256 + tid;
      int jj = flat >> 6, b = flat & 63;
      int jg = jbase + jj;
      float r  = sigmoid_f(g_r[b * GSTR + jj] + bsum_r[jj]);
      float zz = sigmoid_f(g_z[b * GSTR + jj] + bsum_z[jj]);
      float n  = tanhf(g_nx[b * GSTR + jj] + bx_n[jj] +
                       r * (g_nh[b * GSTR + jj] + bh_n[jj]));
      float hold = hc[hidx(jg, b)];
      hn[hidx(jg, b)] = (1.0f - zz) * n + zz * hold;
    }
    grid_barrier(cnt, (unsigned int)(NWG * (t + 1)));

    // out_t = h_new @ out_W^T + out_b  (computed redundantly in every WG
    // so the feedback input is local; avoids a 2nd device barrier)
    v8f c0 = {}, c1 = {};
#pragma unroll 4
    for (int s = 0; s < HIDDEN_D / 4; ++s) {
      int p = 2 * s + hi;
      v2f a  = *(const v2f*)(hn + (p * BATCH_D + mt2 * 16 + lo) * 2);
      v2f b0 = *(const v2f*)(outWswz + (p * 64 + nc + lo) * 2);
      v2f b1 = *(const v2f*)(outWswz + (p * 64 + nc + 16 + lo) * 2);
      c0 = wmma4(a, b0, c0);
      c1 = wmma4(a, b1, c1);
    }
#pragma unroll
    for (int v = 0; v < 8; ++v) {
      int b = mt2 * 16 + v + 8 * hi;
      int i0 = nc + lo, i1 = nc + 16 + lo;
      float o0 = c0[v] + ob_s[i0];
      float o1 = c1[v] + ob_s[i1];
      inpS[(i0 >> 1) * (2 * BATCH_D) + b * 2 + (i0 & 1)] = o0;
      inpS[(i1 >> 1) * (2 * BATCH_D) + b * 2 + (i1 & 1)] = o1;
      if (g == 0) {
        dout[((size_t)b * T_STEPS + t) * INPUT_D + i0] = o0;
        dout[((size_t)b * T_STEPS + t) * INPUT_D + i1] = o1;
      }
    }
    __syncthreads();   // inpS ready before next step's gate GEMM
  }
}

// ---------------- host launcher ----------------
extern "C" void kernel_launch(void* const* d_in, const int* in_sizes, int n_in,
                              void* d_out, int out_size, void* d_ws, size_t ws_size,
                              hipStream_t stream) {
  const float* x    = (const float*)d_in[0];
  const float* eps  = (const float*)d_in[1];
  const float* eWih = (const float*)d_in[2];
  const float* eWhh = (const float*)d_in[3];
  const float* eBih = (const float*)d_in[4];
  const float* eBhh = (const float*)d_in[5];
  const float* muW  = (const float*)d_in[6];
  const float* muB  = (const float*)d_in[7];
  const float* lvW  = (const float*)d_in[8];
  const float* lvB  = (const float*)d_in[9];
  const float* l2hW = (const float*)d_in[10];
  const float* l2hB = (const float*)d_in[11];
  const float* dWih = (const float*)d_in[12];
  const float* dWhh = (const float*)d_in[13];
  const float* dBih = (const float*)d_in[14];
  const float* dBhh = (const float*)d_in[15];
  const float* outW = (const float*)d_in[16];
  const float* outB = (const float*)d_in[17];
  float* out = (float*)d_out;

  float* ws = (float*)d_ws;
  float* hbuf0   = ws;                  // 32768 floats (pair-interleaved h, buf0)
  float* hbuf1   = ws + 32768;          // 32768 floats (buf1)
  float* zbuf    = ws + 65536;          // 8192 floats
  float* outWswz = ws + 73728;          // 32768 floats
  unsigned int* cnt = (unsigned int*)(ws + 106496);  // [0]=enc, [1]=dec

  prep_kernel<<<dim3(128), dim3(256), 0, stream>>>(outW, hbuf0, outWswz, cnt);

  size_t smem_enc = (size_t)((HIDDEN_D / 2) * NSTR + (INPUT_D / 2) * NSTR +
                             4 * BATCH_D * GSTR + 4 * CHUNK) * sizeof(float);
  enc_kernel<<<dim3(NWG), dim3(256), smem_enc, stream>>>(
      x, eWih, eWhh, eBih, eBhh, hbuf0, hbuf1, cnt + 0);

  latent_kernel<<<dim3(32), dim3(256), 0, stream>>>(
      eps, muW, muB, lvW, lvB, hbuf0, zbuf, out);

  l2h_kernel<<<dim3(128), dim3(256), 0, stream>>>(l2hW, l2hB, zbuf, hbuf0);

  size_t smem_dec = smem_enc +
                    (size_t)((INPUT_D / 2) * BATCH_D * 2 + INPUT_D) * sizeof(float);
  dec_kernel<<<dim3(NWG), dim3(256), smem_dec, stream>>>(
      dWih, dWhh, dBih, dBhh, outB, outWswz, hbuf0, hbuf1, out, cnt + 1);
}